// CCQC_classifier_31507880083982
// MI455X (gfx1250) — compile-verified
//
#include <hip/hip_runtime.h>
#include <math.h>

#define NQ    10
#define DIM   1024        // 2^10 state size
#define BATCH 8192
#define DEPTH 5

typedef float v2f __attribute__((ext_vector_type(2)));
typedef float v8f __attribute__((ext_vector_type(8)));

// ---------------------------------------------------------------------------
// Kernel 0: zero the logit accumulators and the scalar output.
// ---------------------------------------------------------------------------
__global__ void k_zero(float* __restrict__ logits, float* __restrict__ out) {
    int i = blockIdx.x * blockDim.x + threadIdx.x;
    if (i < BATCH * 2) logits[i] = 0.0f;
    if (i == 0) out[0] = 0.0f;
}

// ---------------------------------------------------------------------------
// Kernel 1: pad 784 -> 1024 and L2-normalize each sample (real amplitudes).
// ---------------------------------------------------------------------------
__global__ void k_init_state(const float* __restrict__ x, float* __restrict__ A) {
    __shared__ float red[256];
    const int b   = blockIdx.x;
    const int tid = threadIdx.x;
    const float* xr = x + (size_t)b * 784;

    float ss = 0.0f;
    for (int k = tid; k < 784; k += 256) { float v = xr[k]; ss += v * v; }
    red[tid] = ss;
    __syncthreads();
    for (int off = 128; off > 0; off >>= 1) {
        if (tid < off) red[tid] += red[tid + off];
        __syncthreads();
    }
    const float rn = rsqrtf(red[0]);
    float* Ar = A + (size_t)b * DIM;
    for (int k = tid; k < DIM; k += 256)
        Ar[k] = (k < 784) ? xr[k] * rn : 0.0f;
}

// ---------------------------------------------------------------------------
// Kernel 2: build the circuit unitary, one column (basis state) per block.
// State lives in LDS (2 x 4KB). Qubit q acts on index bit (9 - q).
// ---------------------------------------------------------------------------
__device__ __forceinline__ int ins1(int x, int pos) {
    return ((x >> pos) << (pos + 1)) | (x & ((1 << pos) - 1));
}

// RX(t): [[c, -i s],[-i s, c]], c=cos(t/2), s=sin(t/2)
__device__ __forceinline__ void gate_rx(float* sr, float* si, int bit, float ht, int tid) {
    const float c = cosf(ht), s = sinf(ht);
    for (int p = tid; p < DIM / 2; p += 256) {
        const int i0 = ins1(p, bit), i1 = i0 | (1 << bit);
        const float ar = sr[i0], ai = si[i0], br = sr[i1], bi = si[i1];
        sr[i0] = c * ar + s * bi;  si[i0] = c * ai - s * br;
        sr[i1] = c * br + s * ai;  si[i1] = c * bi - s * ar;
    }
    __syncthreads();
}

// RZ(t): diag(e^{-it/2}, e^{+it/2})
__device__ __forceinline__ void gate_rz(float* sr, float* si, int bit, float ht, int tid) {
    const float c = cosf(ht), s = sinf(ht);
    for (int p = tid; p < DIM / 2; p += 256) {
        const int i0 = ins1(p, bit), i1 = i0 | (1 << bit);
        const float ar = sr[i0], ai = si[i0], br = sr[i1], bi = si[i1];
        sr[i0] = c * ar + s * ai;  si[i0] = c * ai - s * ar;
        sr[i1] = c * br - s * bi;  si[i1] = c * bi + s * br;
    }
    __syncthreads();
}

// Controlled RX/RZ: apply on target-bit pairs where ctrl bit == 1.
__device__ __forceinline__ void gate_c(float* sr, float* si, int cb, int tb,
                                       float ht, int tid, int is_rz) {
    const float c = cosf(ht), s = sinf(ht);
    const int lo = cb < tb ? cb : tb;
    const int hi = cb < tb ? tb : cb;
    for (int p = tid; p < DIM / 4; p += 256) {
        const int base = ins1(ins1(p, lo), hi) | (1 << cb);
        const int i0 = base, i1 = base | (1 << tb);
        const float ar = sr[i0], ai = si[i0], br = sr[i1], bi = si[i1];
        if (is_rz) {
            sr[i0] = c * ar + s * ai;  si[i0] = c * ai - s * ar;
            sr[i1] = c * br - s * bi;  si[i1] = c * bi + s * br;
        } else {
            sr[i0] = c * ar + s * bi;  si[i0] = c * ai - s * br;
            sr[i1] = c * br + s * ai;  si[i1] = c * bi - s * ar;
        }
    }
    __syncthreads();
}

__global__ void k_unitary(const float* __restrict__ w,   // (5,10,5)
                          const float* __restrict__ w1,  // (5)
                          const float* __restrict__ w2,  // (5)
                          float* __restrict__ Utr,       // Ut[k][n] = Re U[n][k]
                          float* __restrict__ Uti) {
    __shared__ float sr[DIM];
    __shared__ float si[DIM];
    const int col = blockIdx.x;   // basis state / column of U == row of Ut
    const int tid = threadIdx.x;

    for (int i = tid; i < DIM; i += 256) { sr[i] = (i == col) ? 1.0f : 0.0f; si[i] = 0.0f; }
    __syncthreads();

    for (int d = 0; d < DEPTH; ++d) {
        for (int q = 0; q < NQ; ++q) {
            const int bit = 9 - q;
            const float* wq = w + (size_t)(d * NQ + q) * 5;
            gate_rx(sr, si, bit, wq[0] * 0.5f, tid);
            gate_rz(sr, si, bit, wq[1] * 0.5f, tid);
            gate_rx(sr, si, bit, wq[2] * 0.5f, tid);
        }
        const int r = (d & 1) ? 3 : 1;
        for (int q = 0; q < NQ; ++q) {
            const int cq = (q + r) % NQ;
            const int tb = 9 - q, cb = 9 - cq;
            const float* wq = w + (size_t)(d * NQ + q) * 5;
            gate_c(sr, si, cb, tb, wq[3] * 0.5f, tid, 1);  // CRZ
            gate_c(sr, si, cb, tb, wq[4] * 0.5f, tid, 0);  // CRX
        }
        gate_rx(sr, si, 9, w1[d] * 0.5f, tid);
        gate_rz(sr, si, 9, w2[d] * 0.5f, tid);
    }

    for (int i = tid; i < DIM; i += 256) {
        Utr[(size_t)col * DIM + i] = sr[i];
        Uti[(size_t)col * DIM + i] = si[i];
    }
}

// ---------------------------------------------------------------------------
// Kernel 3: fused WMMA GEMM + |.|^2 + Z-sign epilogue.
// One wave (32 threads) computes a 64(M) x 16(N) strip of S' = A @ Ut,
// for both the real (Utr) and imag (Uti) parts, K = 1024 in steps of 4.
// Epilogue reduces prob*sign over the 16 N-columns and atomically adds
// into per-sample logits.
// ---------------------------------------------------------------------------
__global__ void k_gemm_logits(const float* __restrict__ A,
                              const float* __restrict__ Utr,
                              const float* __restrict__ Uti,
                              float* __restrict__ logits) {
    const int lane = threadIdx.x;       // 0..31 (one wave32 per block)
    const int hf   = lane >> 4;         // lane half: selects K pair
    const int l15  = lane & 15;
    const int ntile = blockIdx.x;       // 0..63  (N = state index / 16)
    const int mblk  = blockIdx.y;       // 0..127 (M = sample / 64)
    const int n = ntile * 16 + l15;     // global state index (B frag col, D col)
    const int mbase = mblk * 64;

    v8f accr[4], acci[4];
    #pragma unroll
    for (int s = 0; s < 4; ++s) { accr[s] = {}; acci[s] = {}; }

    for (int k0 = 0; k0 < DIM; k0 += 4) {
        const int ka = k0 + hf * 2;     // this lane's K pair
        v2f bfr, bfi;
        bfr.x = Utr[(size_t)ka * DIM + n];
        bfr.y = Utr[(size_t)(ka + 1) * DIM + n];
        bfi.x = Uti[(size_t)ka * DIM + n];
        bfi.y = Uti[(size_t)(ka + 1) * DIM + n];
        #pragma unroll
        for (int s = 0; s < 4; ++s) {
            const int m = mbase + s * 16 + l15;          // A frag row
            const float* ap = A + (size_t)m * DIM + ka;
            v2f af; af.x = ap[0]; af.y = ap[1];
            accr[s] = __builtin_amdgcn_wmma_f32_16x16x4_f32(
                false, af, false, bfr, (short)0, accr[s], false, false);
            acci[s] = __builtin_amdgcn_wmma_f32_16x16x4_f32(
                false, af, false, bfi, (short)0, acci[s], false, false);
        }
    }

    // Z-expectation signs for logits 0 (qubit 0 -> bit 9) and 1 (bit 8)
    const float z0 = ((n >> 9) & 1) ? -1.0f : 1.0f;
    const float z1 = ((n >> 8) & 1) ? -1.0f : 1.0f;

    #pragma unroll
    for (int s = 0; s < 4; ++s) {
        #pragma unroll
        for (int v = 0; v < 8; ++v) {
            const float re = accr[s][v], im = acci[s][v];
            float w0 = (re * re + im * im) * z0;
            float w1 = (re * re + im * im) * z1;
            // reduce over the 16 lanes sharing this row (xor<16 stays in half)
            #pragma unroll
            for (int off = 1; off < 16; off <<= 1) {
                w0 += __shfl_xor(w0, off, 32);
                w1 += __shfl_xor(w1, off, 32);
            }
            if (l15 == 0) {
                const int m = mbase + s * 16 + hf * 8 + v;   // sample index
                atomicAdd(&logits[2 * m + 0], w0);
                atomicAdd(&logits[2 * m + 1], w1);
            }
        }
    }
}

// ---------------------------------------------------------------------------
// Kernel 4: per-sample NLL + mean reduction into the scalar output.
// ---------------------------------------------------------------------------
__global__ void k_loss(const float* __restrict__ logits,
                       const int* __restrict__ y,
                       float* __restrict__ out) {
    __shared__ float red[256];
    const int b = blockIdx.x * blockDim.x + threadIdx.x;
    float nll = 0.0f;
    if (b < BATCH) {
        const float l0 = logits[2 * b], l1 = logits[2 * b + 1];
        const float mx = fmaxf(l0, l1);
        const float lse = mx + logf(expf(l0 - mx) + expf(l1 - mx));
        const float ly = (y[b] != 0) ? l1 : l0;
        nll = lse - ly;
    }
    red[threadIdx.x] = nll;
    __syncthreads();
    for (int off = 128; off > 0; off >>= 1) {
        if (threadIdx.x < off) red[threadIdx.x] += red[threadIdx.x + off];
        __syncthreads();
    }
    if (threadIdx.x == 0) atomicAdd(out, red[0] * (1.0f / BATCH));
}

// ---------------------------------------------------------------------------
extern "C" void kernel_launch(void* const* d_in, const int* in_sizes, int n_in,
                              void* d_out, int out_size, void* d_ws, size_t ws_size,
                              hipStream_t stream) {
    const float* x  = (const float*)d_in[0];   // (8192, 784)
    const int*   y  = (const int*)  d_in[1];   // (8192,)
    const float* w  = (const float*)d_in[2];   // (5,10,5)
    const float* w1 = (const float*)d_in[3];   // (5,)
    const float* w2 = (const float*)d_in[4];   // (5,)
    float* out = (float*)d_out;

    char* ws = (char*)d_ws;
    const size_t szA  = (size_t)BATCH * DIM * sizeof(float);   // 32 MB
    const size_t szU  = (size_t)DIM * DIM * sizeof(float);     //  4 MB each
    float* A      = (float*)(ws);
    float* Utr    = (float*)(ws + szA);
    float* Uti    = (float*)(ws + szA + szU);
    float* logits = (float*)(ws + szA + 2 * szU);              // 8192 x 2

    k_zero<<<(BATCH * 2 + 255) / 256, 256, 0, stream>>>(logits, out);
    k_init_state<<<BATCH, 256, 0, stream>>>(x, A);
    k_unitary<<<DIM, 256, 0, stream>>>(w, w1, w2, Utr, Uti);
    dim3 gg(DIM / 16, BATCH / 64);
    k_gemm_logits<<<gg, 32, 0, stream>>>(A, Utr, Uti, logits);
    k_loss<<<BATCH / 256, 256, 0, stream>>>(logits, y, out);
}